// GCNEncoder2_63754494542545
// MI455X (gfx1250) — compile-verified
//
#include <hip/hip_runtime.h>

// GCN 2-layer forward for MI455X (gfx1250, wave32).
//   h1 = x @ W1                      -> WMMA f32 16x16x4 GEMM
//   a1[dst] += w * h1[src]           -> wave-per-edge, native fp32 atomics (L2-resident)
//   h2 = a1 @ W2                     -> WMMA f32 16x16x4 GEMM
//   out[dst] += w * h2[src]          -> wave-per-edge atomics into d_out

typedef __attribute__((ext_vector_type(2))) float v2f;
typedef __attribute__((ext_vector_type(8))) float v8f;

#define N_NODES 100000
#define N_EDGES 1600000
#define OUTC 32   // both layers produce 32 features

// ---------------------------------------------------------------------------
// GEMM: H[n_rows x 32] = X[n_rows x K] @ W[K x 32], fp32 via V_WMMA_F32_16X16X4_F32.
// One wave per 16-row tile; two 16x16 accumulators cover the 32 output columns.
// A 16x4 layout: lanes 0-15 -> K={kb,kb+1}, lanes 16-31 -> K={kb+2,kb+3} (m = lane&15).
// B 4x16 layout mirrors A with N across lanes. C/D: VGPR r, lanes 0-15 -> M=r,
// lanes 16-31 -> M=8+r, N = lane&15.
// ---------------------------------------------------------------------------
template<int K>
__global__ __launch_bounds__(256) void gemm_x32_wmma(
    const float* __restrict__ X,
    const float* __restrict__ W,   // K x 32 row-major
    float* __restrict__ H,         // n_rows x 32
    int n_rows)
{
  const int lane = threadIdx.x & 31;
  const int tile = (int)(blockIdx.x * (blockDim.x >> 5) + (threadIdx.x >> 5));
  const int n_tiles = n_rows >> 4;            // 100000/16 = 6250, exact
  if (tile >= n_tiles) return;                // wave-uniform: EXEC stays all-1s

  const int  m   = lane & 15;                 // row (A) / col (B,D) within 16
  const int  kh  = lane >> 4;                 // which K-pair this half-wave holds
  const long row0 = (long)tile << 4;

  const float* xrow = X + (row0 + m) * K;

  v8f c0 = {};  // columns 0..15
  v8f c1 = {};  // columns 16..31

  #pragma unroll 8
  for (int kb = 0; kb < K; kb += 4) {
    const int ka = kb + 2 * kh;
    v2f a;
    a.x = xrow[ka];
    a.y = xrow[ka + 1];
    v2f b0, b1;
    b0.x = W[ka * OUTC + m];
    b0.y = W[(ka + 1) * OUTC + m];
    b1.x = W[ka * OUTC + 16 + m];
    b1.y = W[(ka + 1) * OUTC + 16 + m];
    c0 = __builtin_amdgcn_wmma_f32_16x16x4_f32(false, a, false, b0, (short)0, c0, false, false);
    c1 = __builtin_amdgcn_wmma_f32_16x16x4_f32(false, a, false, b1, (short)0, c1, false, false);
  }

  // Store: this lane owns rows row0 + 8*kh + r (r=0..7), column m (and m+16).
  float* hrow = H + (row0 + 8 * kh) * OUTC + m;
  #pragma unroll
  for (int r = 0; r < 8; ++r) {
    hrow[r * OUTC]      = c0[r];
    hrow[r * OUTC + 16] = c1[r];
  }
}

// ---------------------------------------------------------------------------
// Native fp32 global atomic add, no return (tracked on STOREcnt; drained by the
// implicit wait-idle at s_endpgm). scope:SCOPE_DEV forces the RMW to execute at
// the device-coherent cache level (L2) so cross-WGP accumulation is correct.
// Pinned via inline asm so LLVM cannot expand it into a CAS loop.
// ---------------------------------------------------------------------------
__device__ __forceinline__ void atomic_add_f32_dev(float* p, float v) {
  asm volatile("global_atomic_add_f32 %0, %1, off scope:SCOPE_DEV"
               :
               : "v"(p), "v"(v)
               : "memory");
}

// ---------------------------------------------------------------------------
// Edge scatter: out[dst[e]][f] += w[e] * H[src[e]][f].
// One wave per edge (grid-stride): 32 lanes = 32 features, 128 B coalesced
// gather from H (L2-resident, 12.8 MB) and 32 fp32 atomics into out (L2-resident).
// eidx is [2, E]: row 0 = src, row 1 = dst (int32; JAX x64-off downcasts int64).
// ---------------------------------------------------------------------------
__global__ __launch_bounds__(256) void gcn_scatter(
    const float* __restrict__ H,
    const int*   __restrict__ eidx,
    const float* __restrict__ ew,
    float*       __restrict__ out,
    int n_edges)
{
  const int lane   = threadIdx.x & 31;
  const int wave0  = (int)((blockIdx.x * (long)blockDim.x + threadIdx.x) >> 5);
  const int nwaves = (int)(((long)gridDim.x * blockDim.x) >> 5);

  for (int e = wave0; e < n_edges; e += nwaves) {
    const int   s = eidx[e];
    const int   d = eidx[n_edges + e];
    const float w = ew[e];
    const float v = w * H[(long)s * OUTC + lane];
    atomic_add_f32_dev(&out[(long)d * OUTC + lane], v);
  }
}

// ---------------------------------------------------------------------------
extern "C" void kernel_launch(void* const* d_in, const int* in_sizes, int n_in,
                              void* d_out, int out_size, void* d_ws, size_t ws_size,
                              hipStream_t stream) {
  const float* x  = (const float*)d_in[0];   // [100000, 128] fp32
  const float* W1 = (const float*)d_in[1];   // [128, 32]     fp32
  const float* W2 = (const float*)d_in[2];   // [32, 32]      fp32
  const float* ew = (const float*)d_in[3];   // [1600000]     fp32
  const int*   ei = (const int*)d_in[4];     // [2, 1600000]  int32
  float* out = (float*)d_out;                // [100000, 32]  fp32

  const size_t feat_bytes = (size_t)N_NODES * OUTC * sizeof(float);  // 12.8 MB
  float* h    = (float*)d_ws;                           // GEMM output (reused for both layers)
  float* agg1 = (float*)((char*)d_ws + feat_bytes);     // layer-1 aggregation

  // Scatter targets must start at zero every call (deterministic; d_out is poisoned).
  hipMemsetAsync(agg1, 0, feat_bytes, stream);
  hipMemsetAsync(out,  0, feat_bytes, stream);

  const int tiles = N_NODES / 16;                 // 6250
  const int wpb   = 256 / 32;                     // 8 waves per block
  dim3 gemm_grid((tiles + wpb - 1) / wpb);        // 782 blocks

  // Layer 1
  gemm_x32_wmma<128><<<gemm_grid, 256, 0, stream>>>(x, W1, h, N_NODES);
  gcn_scatter<<<4096, 256, 0, stream>>>(h, ei, ew, agg1, N_EDGES);

  // Layer 2
  gemm_x32_wmma<32><<<gemm_grid, 256, 0, stream>>>(agg1, W2, h, N_NODES);
  gcn_scatter<<<4096, 256, 0, stream>>>(h, ei, ew, out, N_EDGES);
}